// GraphConvPoolNNProtein_18305150616266
// MI455X (gfx1250) — compile-verified
//
#include <hip/hip_runtime.h>
#include <cstdint>
#include <cstddef>

#define N_NODES 250000
#define N_EDGES 2000000
#define HDIM 32
#define NGRAPHS 128
#define HASH_BITS 22
#define HASH_SIZE (1u << HASH_BITS)
#define HASH_MASK (HASH_SIZE - 1u)
#define CC_ITERS 34
#define TB 256

typedef __attribute__((ext_vector_type(2))) float v2f;
typedef __attribute__((ext_vector_type(8))) float v8f;

__device__ __forceinline__ float sigmoidf(float z) { return 1.0f / (1.0f + expf(-z)); }

// ---------------- fill kernels (graph-capture-safe init) ----------------
__global__ void k_fill_f32(float* p, float v, int n) {
  int i = blockIdx.x * blockDim.x + threadIdx.x;
  if (i < n) p[i] = v;
}
__global__ void k_fill_u64(unsigned long long* p, unsigned long long v, int n) {
  int i = blockIdx.x * blockDim.x + threadIdx.x;
  if (i < n) p[i] = v;
}
__global__ void k_fill_i32(int* p, int v, int n) {
  int i = blockIdx.x * blockDim.x + threadIdx.x;
  if (i < n) p[i] = v;
}

// ---------------- GCN-1 degree / norm ----------------
__global__ void k_deg1(const int* __restrict__ el, float* __restrict__ deg, int nE) {
  int e = blockIdx.x * blockDim.x + threadIdx.x;
  if (e < nE) atomicAdd(&deg[el[2 * e + 1]], 1.0f);
}
__global__ void k_dis1(float* deg, int n) {
  int i = blockIdx.x * blockDim.x + threadIdx.x;
  if (i < n) deg[i] = rsqrtf(deg[i] + 1.0f); // deg + self_w(=1) >= 1 always
}

// ---------------- WMMA GEMM: out[N,32] = x[N,3] @ W1[3,32]  (K padded to 4) ----------------
// Branch-free K padding: clamp index in-bounds, multiply value by 0/1 mask.
__global__ void k_gemm_in3_wmma(const float* __restrict__ x, const float* __restrict__ W1,
                                float* __restrict__ out, int nrows) {
  int lane = threadIdx.x & 31;
  int wave = blockIdx.x * (blockDim.x >> 5) + (threadIdx.x >> 5);
  int row0 = wave * 16;
  if (row0 + 16 > nrows) return;           // wave-uniform guard: EXEC stays all-ones
  int r = lane & 15;
  int kk = (lane >> 4) * 2;                // lanes 0-15 -> K=0,1 ; lanes 16-31 -> K=2,3
  int row = row0 + r;

  bool hi_valid = (kk + 1) < 3;            // K=3 is zero-pad
  int k1 = hi_valid ? (kk + 1) : 0;        // always in-bounds
  float mk = hi_valid ? 1.0f : 0.0f;

  v2f a;
  a.x = x[row * 3 + kk];
  a.y = x[row * 3 + k1] * mk;
  v2f b0, b1;
  b0.x = W1[kk * HDIM + r];
  b0.y = W1[k1 * HDIM + r] * mk;
  b1.x = W1[kk * HDIM + 16 + r];
  b1.y = W1[k1 * HDIM + 16 + r] * mk;

  v8f acc0 = {}; v8f acc1 = {};
  acc0 = __builtin_amdgcn_wmma_f32_16x16x4_f32(false, a, false, b0, (short)0, acc0, false, false);
  acc1 = __builtin_amdgcn_wmma_f32_16x16x4_f32(false, a, false, b1, (short)0, acc1, false, false);

  int mbase = row0 + ((lane >> 4) << 3);   // lanes 16-31 hold rows M=8..15
#pragma unroll
  for (int t = 0; t < 8; ++t) {
    out[(mbase + t) * HDIM + r]      = acc0[t];
    out[(mbase + t) * HDIM + 16 + r] = acc1[t];
  }
}

// ---------------- WMMA GEMM: out[N,32] = A[N,32] @ W[32,32] ----------------
// W staged once per block into LDS, transposed so B fragments are ds_load_b64 pairs.
__global__ void k_gemm_h32_wmma(const float* __restrict__ A, const float* __restrict__ W,
                                float* __restrict__ out, int nrows) {
  __shared__ float WT[HDIM * HDIM];        // WT[n*32 + k] = W[k*32 + n]
  int tid = threadIdx.x;
  {
    const float4* W4 = (const float4*)W;   // 256 threads * 4 floats = 1024
    float4 w = W4[tid];
    int m = tid * 4;                       // linear index into row-major W[k][n]
    int k = m >> 5;
    int n0 = m & 31;
    WT[(n0 + 0) * HDIM + k] = w.x;
    WT[(n0 + 1) * HDIM + k] = w.y;
    WT[(n0 + 2) * HDIM + k] = w.z;
    WT[(n0 + 3) * HDIM + k] = w.w;
  }
  __syncthreads();                          // all waves participate before any exits

  int lane = tid & 31;
  int wave = blockIdx.x * (blockDim.x >> 5) + (tid >> 5);
  int row0 = wave * 16;
  if (row0 + 16 > nrows) return;           // wave-uniform
  int r = lane & 15;
  int khalf = (lane >> 4) * 2;
  int row = row0 + r;

  v8f acc0 = {}; v8f acc1 = {};
#pragma unroll
  for (int k0 = 0; k0 < HDIM; k0 += 4) {
    int k = k0 + khalf;                     // even -> 8B aligned everywhere
    v2f a  = *(const v2f*)&A[row * HDIM + k];
    v2f b0 = *(const v2f*)&WT[r * HDIM + k];
    v2f b1 = *(const v2f*)&WT[(16 + r) * HDIM + k];
    acc0 = __builtin_amdgcn_wmma_f32_16x16x4_f32(false, a, false, b0, (short)0, acc0, false, false);
    acc1 = __builtin_amdgcn_wmma_f32_16x16x4_f32(false, a, false, b1, (short)0, acc1, false, false);
  }
  int mbase = row0 + ((lane >> 4) << 3);
#pragma unroll
  for (int t = 0; t < 8; ++t) {
    out[(mbase + t) * HDIM + r]      = acc0[t];
    out[(mbase + t) * HDIM + 16 + r] = acc1[t];
  }
}

// ---------------- GCN-1 message + combine ----------------
__global__ void k_msg1(const int* __restrict__ el, const float* __restrict__ xw,
                       const float* __restrict__ dis, float* __restrict__ msg, int nE) {
  int e = blockIdx.x * blockDim.x + threadIdx.x;
  if (e >= nE) return;
  int s = el[2 * e], d = el[2 * e + 1];
  float c = dis[s] * dis[d];
  const float4* hs = (const float4*)(xw + (size_t)s * HDIM);
  float* md = msg + (size_t)d * HDIM;
#pragma unroll
  for (int q = 0; q < HDIM / 4; ++q) {
    float4 v = hs[q];
    atomicAdd(&md[q * 4 + 0], v.x * c);
    atomicAdd(&md[q * 4 + 1], v.y * c);
    atomicAdd(&md[q * 4 + 2], v.z * c);
    atomicAdd(&md[q * 4 + 3], v.w * c);
  }
}
__global__ void k_relu1(const float4* __restrict__ msg, const float4* __restrict__ xw,
                        const float* __restrict__ dis, const float4* __restrict__ b1,
                        float4* __restrict__ x1, int n) {
  int idx = blockIdx.x * blockDim.x + threadIdx.x;   // over n * 8 float4 chunks
  if (idx >= n * (HDIM / 4)) return;
  int i = idx >> 3, h4 = idx & 7;
  float di = dis[i];
  float s = di * di;
  float4 m = msg[idx], w = xw[idx], b = b1[h4];
  float4 o;
  o.x = fmaxf(m.x + w.x * s + b.x, 0.0f);
  o.y = fmaxf(m.y + w.y * s + b.y, 0.0f);
  o.z = fmaxf(m.z + w.z * s + b.z, 0.0f);
  o.w = fmaxf(m.w + w.w * s + b.w, 0.0f);
  x1[idx] = o;
}

// ---------------- edge scores / selection ----------------
__global__ void k_score(const int* __restrict__ el, const float* __restrict__ x1,
                        const float* __restrict__ wpool, const float* __restrict__ bpool,
                        float* __restrict__ score, int* __restrict__ u, int* __restrict__ v, int nE) {
  int e = blockIdx.x * blockDim.x + threadIdx.x;
  if (e >= nE) return;
  int s = el[2 * e], d = el[2 * e + 1];
  const float4* hs = (const float4*)(x1 + (size_t)s * HDIM);
  const float4* hd = (const float4*)(x1 + (size_t)d * HDIM);
  const float4* wp = (const float4*)wpool;
  float acc = bpool[0];
#pragma unroll
  for (int q = 0; q < HDIM / 4; ++q) {
    float4 a = hs[q], b = hd[q], wa = wp[q], wb = wp[HDIM / 4 + q];
    acc += a.x * wa.x + a.y * wa.y + a.z * wa.z + a.w * wa.w;
    acc += b.x * wb.x + b.y * wb.y + b.z * wb.z + b.w * wb.w;
  }
  float sc = sigmoidf(acc);
  score[e] = sc;
  bool sel = sc > 0.5f;
  u[e] = sel ? s : 0;
  v[e] = sel ? d : 0;
}

// ---------------- connected components (min-label + pointer jumping) ----------------
__global__ void k_lab_init(int* lab, int n) {
  int i = blockIdx.x * blockDim.x + threadIdx.x;
  if (i < n) lab[i] = i;
}
__global__ void k_cc_relax(const int* __restrict__ u, const int* __restrict__ v,
                           int* __restrict__ lab, int nE) {
  int e = blockIdx.x * blockDim.x + threadIdx.x;
  if (e >= nE) return;
  int ue = u[e], ve = v[e];
  int a = lab[ue], b = lab[ve];
  int m = min(a, b);
  if (a > m) atomicMin(&lab[ue], m);
  if (b > m) atomicMin(&lab[ve], m);
}
__global__ void k_cc_jump(int* lab, int n) {
  int i = blockIdx.x * blockDim.x + threadIdx.x;
  if (i >= n) return;
  int l = lab[i];
  int ll = lab[l];
  if (ll < l) lab[i] = ll;   // monotone decreasing -> race-safe, unique fixpoint
}

// ---------------- cluster score ----------------
__global__ void k_group(const int* __restrict__ el, const int* __restrict__ lab,
                        const float* __restrict__ score, float* __restrict__ cnt,
                        float* __restrict__ ssum, int nE) {
  int e = blockIdx.x * blockDim.x + threadIdx.x;
  if (e >= nE) return;
  float sc = score[e];
  if (sc > 0.5f) {
    int g = lab[el[2 * e]];
    atomicAdd(&cnt[g], 1.0f);
    atomicAdd(&ssum[g], sc);
  }
}
__global__ void k_cscore(float* cnt, const float* ssum, int n) {
  int i = blockIdx.x * blockDim.x + threadIdx.x;
  if (i >= n) return;
  float c = cnt[i];
  cnt[i] = (c > 0.0f) ? (ssum[i] / fmaxf(c, 1.0f)) : 1.0f;  // cnt becomes cscore
}

// ---------------- contracted node features ----------------
__global__ void k_newx_acc(const float* __restrict__ x1, const int* __restrict__ lab,
                           float* __restrict__ newx, int n) {
  int i = blockIdx.x * blockDim.x + threadIdx.x;
  if (i >= n) return;
  int g = lab[i];
  const float4* xi = (const float4*)(x1 + (size_t)i * HDIM);
  float* ng = newx + (size_t)g * HDIM;
#pragma unroll
  for (int q = 0; q < HDIM / 4; ++q) {
    float4 v = xi[q];
    atomicAdd(&ng[q * 4 + 0], v.x);
    atomicAdd(&ng[q * 4 + 1], v.y);
    atomicAdd(&ng[q * 4 + 2], v.z);
    atomicAdd(&ng[q * 4 + 3], v.w);
  }
}
__global__ void k_newx_scale(float4* newx, const float* cscore, int n) {
  int idx = blockIdx.x * blockDim.x + threadIdx.x;   // over n * 8 float4 chunks
  if (idx >= n * (HDIM / 4)) return;
  float c = cscore[idx >> 3];
  float4 v = newx[idx];
  v.x *= c; v.y *= c; v.z *= c; v.w *= c;
  newx[idx] = v;
}

// ---------------- contracted edge dedup (hash) + deg2 ----------------
__global__ void k_contract(const int* __restrict__ el, const int* __restrict__ lab,
                           unsigned long long* __restrict__ hash,
                           unsigned long long* __restrict__ cedges, int* __restrict__ count,
                           float* __restrict__ deg2, int nE) {
  int e = blockIdx.x * blockDim.x + threadIdx.x;
  if (e >= nE) return;
  int ru = lab[el[2 * e]];
  int rv = lab[el[2 * e + 1]];
  if (ru == rv) return;                       // w = first & (ru != rv)
  unsigned long long key = ((unsigned long long)(unsigned)ru << 32) | (unsigned)rv;
  unsigned int slot = (unsigned int)((key * 0x9E3779B97F4A7C15ull) >> 40) & HASH_MASK;
  for (;;) {
    unsigned long long prev = atomicCAS(&hash[slot], ~0ull, key);
    if (prev == ~0ull) {                       // this edge is the unique contributor
      int pos = atomicAdd(count, 1);
      cedges[pos] = key;
      atomicAdd(&deg2[rv], 1.0f);
      return;
    }
    if (prev == key) return;                   // duplicate
    slot = (slot + 1) & HASH_MASK;
  }
}
__global__ void k_dis2(float* deg2, const int* lab, int n) {
  int i = blockIdx.x * blockDim.x + threadIdx.x;
  if (i >= n) return;
  float d = deg2[i] + ((lab[i] == i) ? 1.0f : 0.0f);   // self_w = repf
  deg2[i] = (d > 0.0f) ? rsqrtf(fmaxf(d, 1e-12f)) : 0.0f;
}

// ---------------- GCN-2 message + combine ----------------
__global__ void k_msg2(const unsigned long long* __restrict__ cedges, const int* __restrict__ count,
                       const float* __restrict__ nxw, const float* __restrict__ dis2,
                       float* __restrict__ msg2) {
  int e = blockIdx.x * blockDim.x + threadIdx.x;
  if (e >= *count) return;
  unsigned long long key = cedges[e];
  int ru = (int)(key >> 32);
  int rv = (int)(key & 0xffffffffull);
  float c = dis2[ru] * dis2[rv];
  const float4* hs = (const float4*)(nxw + (size_t)ru * HDIM);
  float* md = msg2 + (size_t)rv * HDIM;
#pragma unroll
  for (int q = 0; q < HDIM / 4; ++q) {
    float4 v = hs[q];
    atomicAdd(&md[q * 4 + 0], v.x * c);
    atomicAdd(&md[q * 4 + 1], v.y * c);
    atomicAdd(&md[q * 4 + 2], v.z * c);
    atomicAdd(&md[q * 4 + 3], v.w * c);
  }
}
__global__ void k_relu2(const float4* __restrict__ msg2, const float4* __restrict__ nxw,
                        const float* __restrict__ dis2, const int* __restrict__ lab,
                        const float4* __restrict__ b3, float4* __restrict__ x3, int n) {
  int idx = blockIdx.x * blockDim.x + threadIdx.x;   // over n * 8 float4 chunks
  if (idx >= n * (HDIM / 4)) return;
  int i = idx >> 3, h4 = idx & 7;
  float rep = (lab[i] == i) ? 1.0f : 0.0f;
  float di = dis2[i];
  float s = di * di * rep;
  float4 m = msg2[idx], w = nxw[idx], b = b3[h4];
  float4 o;
  o.x = fmaxf(m.x + w.x * s + b.x, 0.0f);
  o.y = fmaxf(m.y + w.y * s + b.y, 0.0f);
  o.z = fmaxf(m.z + w.z * s + b.z, 0.0f);
  o.w = fmaxf(m.w + w.w * s + b.w, 0.0f);
  x3[idx] = o;
}

// ---------------- graph pooling + final MLP ----------------
__global__ void k_pool(const float* __restrict__ x3, const int* __restrict__ lab,
                       const int* __restrict__ batch, float* __restrict__ gs,
                       float* __restrict__ gc, int n) {
  int i = blockIdx.x * blockDim.x + threadIdx.x;
  if (i >= n) return;
  if (lab[i] != i) return;            // x3 * repf gate
  int g = batch[i];
  const float4* xi = (const float4*)(x3 + (size_t)i * HDIM);
#pragma unroll
  for (int q = 0; q < HDIM / 4; ++q) {
    float4 v = xi[q];
    atomicAdd(&gs[g * HDIM + q * 4 + 0], v.x);
    atomicAdd(&gs[g * HDIM + q * 4 + 1], v.y);
    atomicAdd(&gs[g * HDIM + q * 4 + 2], v.z);
    atomicAdd(&gs[g * HDIM + q * 4 + 3], v.w);
  }
  atomicAdd(&gc[g], 1.0f);
}
__global__ void k_final(const float* __restrict__ gs, const float* __restrict__ gc,
                        const float* __restrict__ Wf, const float* __restrict__ bf,
                        float* __restrict__ out) {
  int g = blockIdx.x * blockDim.x + threadIdx.x;
  if (g >= NGRAPHS) return;
  float denom = fmaxf(gc[g], 1.0f);
  float acc = bf[0];
#pragma unroll
  for (int h = 0; h < HDIM; ++h) acc += (gs[g * HDIM + h] / denom) * Wf[h];
  out[g] = sigmoidf(acc);
}

// ======================================================================
extern "C" void kernel_launch(void* const* d_in, const int* in_sizes, int n_in,
                              void* d_out, int out_size, void* d_ws, size_t ws_size,
                              hipStream_t stream) {
  const float* x     = (const float*)d_in[0];
  const int*   el    = (const int*)d_in[1];
  const int*   batch = (const int*)d_in[2];
  const float* W1    = (const float*)d_in[3];
  const float* b1    = (const float*)d_in[4];
  const float* wpool = (const float*)d_in[5];
  const float* bpool = (const float*)d_in[6];
  const float* W3    = (const float*)d_in[7];
  const float* b3    = (const float*)d_in[8];
  const float* Wf    = (const float*)d_in[9];
  const float* bf    = (const float*)d_in[10];
  float* out = (float*)d_out;

  // ---- workspace carve ----
  char* ws = (char*)d_ws;
  auto carve = [&](size_t bytes) -> char* {
    char* p = ws;
    ws += (bytes + 255) & ~(size_t)255;
    return p;
  };
  float* f_xw   = (float*)carve((size_t)N_NODES * HDIM * 4);  // xw, later nxw
  float* f_msg  = (float*)carve((size_t)N_NODES * HDIM * 4);  // msg1, later msg2
  float* f_x1   = (float*)carve((size_t)N_NODES * HDIM * 4);  // x1, later x3
  float* f_newx = (float*)carve((size_t)N_NODES * HDIM * 4);
  float* f_deg  = (float*)carve((size_t)N_NODES * 4);          // deg->dis, deg2->dis2
  int*   i_lab  = (int*)  carve((size_t)N_NODES * 4);
  float* f_cnt  = (float*)carve((size_t)N_NODES * 4);          // -> cscore
  float* f_ssum = (float*)carve((size_t)N_NODES * 4);
  float* f_scr  = (float*)carve((size_t)N_EDGES * 4);
  int*   i_u    = (int*)  carve((size_t)N_EDGES * 4);
  int*   i_v    = (int*)  carve((size_t)N_EDGES * 4);
  unsigned long long* h_tab = (unsigned long long*)carve((size_t)HASH_SIZE * 8);
  unsigned long long* cedge = (unsigned long long*)carve((size_t)N_EDGES * 8);
  int*   i_cnt2 = (int*)  carve(256);
  float* f_gs   = (float*)carve((size_t)NGRAPHS * HDIM * 4);
  float* f_gc   = (float*)carve((size_t)NGRAPHS * 4);

  const int gN   = (N_NODES + TB - 1) / TB;
  const int gNF  = (N_NODES * HDIM + TB - 1) / TB;
  const int gNF4 = (N_NODES * (HDIM / 4) + TB - 1) / TB;
  const int gE   = (N_EDGES + TB - 1) / TB;
  const int gH   = ((int)HASH_SIZE + TB - 1) / TB;
  const int nWaves = N_NODES / 16;                 // 15625 (exact)
  const int gW   = (nWaves + 7) / 8;               // 8 waves / 256-thread block

  // ---- GCN 1 ----
  k_fill_f32<<<gN, TB, 0, stream>>>(f_deg, 0.0f, N_NODES);
  k_deg1<<<gE, TB, 0, stream>>>(el, f_deg, N_EDGES);
  k_dis1<<<gN, TB, 0, stream>>>(f_deg, N_NODES);
  k_gemm_in3_wmma<<<gW, TB, 0, stream>>>(x, W1, f_xw, N_NODES);
  k_fill_f32<<<gNF, TB, 0, stream>>>(f_msg, 0.0f, N_NODES * HDIM);
  k_msg1<<<gE, TB, 0, stream>>>(el, f_xw, f_deg, f_msg, N_EDGES);
  k_relu1<<<gNF4, TB, 0, stream>>>((const float4*)f_msg, (const float4*)f_xw, f_deg,
                                   (const float4*)b1, (float4*)f_x1, N_NODES);

  // ---- edge scores ----
  k_score<<<gE, TB, 0, stream>>>(el, f_x1, wpool, bpool, f_scr, i_u, i_v, N_EDGES);

  // ---- connected components ----
  k_lab_init<<<gN, TB, 0, stream>>>(i_lab, N_NODES);
  for (int it = 0; it < CC_ITERS; ++it) {
    k_cc_relax<<<gE, TB, 0, stream>>>(i_u, i_v, i_lab, N_EDGES);
    k_cc_jump<<<gN, TB, 0, stream>>>(i_lab, N_NODES);
    k_cc_jump<<<gN, TB, 0, stream>>>(i_lab, N_NODES);
  }

  // ---- cluster score ----
  k_fill_f32<<<gN, TB, 0, stream>>>(f_cnt, 0.0f, N_NODES);
  k_fill_f32<<<gN, TB, 0, stream>>>(f_ssum, 0.0f, N_NODES);
  k_group<<<gE, TB, 0, stream>>>(el, i_lab, f_scr, f_cnt, f_ssum, N_EDGES);
  k_cscore<<<gN, TB, 0, stream>>>(f_cnt, f_ssum, N_NODES);

  // ---- contracted features ----
  k_fill_f32<<<gNF, TB, 0, stream>>>(f_newx, 0.0f, N_NODES * HDIM);
  k_newx_acc<<<gN, TB, 0, stream>>>(f_x1, i_lab, f_newx, N_NODES);
  k_newx_scale<<<gNF4, TB, 0, stream>>>((float4*)f_newx, f_cnt, N_NODES);

  // ---- contracted edges (dedup) + deg2 ----
  k_fill_f32<<<gN, TB, 0, stream>>>(f_deg, 0.0f, N_NODES);
  k_fill_u64<<<gH, TB, 0, stream>>>(h_tab, ~0ull, (int)HASH_SIZE);
  k_fill_i32<<<1, 32, 0, stream>>>(i_cnt2, 0, 1);
  k_contract<<<gE, TB, 0, stream>>>(el, i_lab, h_tab, cedge, i_cnt2, f_deg, N_EDGES);
  k_dis2<<<gN, TB, 0, stream>>>(f_deg, i_lab, N_NODES);

  // ---- GCN 2 ----
  k_gemm_h32_wmma<<<gW, TB, 0, stream>>>(f_newx, W3, f_xw, N_NODES);  // nxw into f_xw
  k_fill_f32<<<gNF, TB, 0, stream>>>(f_msg, 0.0f, N_NODES * HDIM);
  k_msg2<<<gE, TB, 0, stream>>>(cedge, i_cnt2, f_xw, f_deg, f_msg);
  k_relu2<<<gNF4, TB, 0, stream>>>((const float4*)f_msg, (const float4*)f_xw, f_deg, i_lab,
                                   (const float4*)b3, (float4*)f_x1, N_NODES);

  // ---- pooling + final ----
  k_fill_f32<<<(NGRAPHS * HDIM + TB - 1) / TB, TB, 0, stream>>>(f_gs, 0.0f, NGRAPHS * HDIM);
  k_fill_f32<<<1, NGRAPHS, 0, stream>>>(f_gc, 0.0f, NGRAPHS);
  k_pool<<<gN, TB, 0, stream>>>(f_x1, i_lab, batch, f_gs, f_gc, N_NODES);
  k_final<<<1, NGRAPHS, 0, stream>>>(f_gs, f_gc, Wf, bf, out);
}